// GroupingPool2d_37323265802444
// MI455X (gfx1250) — compile-verified
//
#include <hip/hip_runtime.h>

namespace {

constexpr int W_IN  = 384;
constexpr int W_OUT = 192;
constexpr int TPB   = 192;            // one thread per output column; 6 waves
constexpr int PAIRS_PER_STAGE = 8;    // row-pairs staged per pipeline step
constexpr int PAIR_FLOATS  = 2 * W_IN;                       // 768 floats (rows 2r, 2r+1)
constexpr int PAIR_BYTES   = PAIR_FLOATS * 4;                // 3072 B
constexpr int STAGE_FLOATS = PAIRS_PER_STAGE * PAIR_FLOATS;  // 6144 floats
constexpr int STAGE_BYTES  = STAGE_FLOATS * 4;               // 24576 B
constexpr float EPS = 1e-6f;

// CDNA5 async data mover: 16 B per lane, global -> LDS, tracked by ASYNCcnt.
// VDST = workgroup-relative LDS byte address (low 32 bits of the flat shared
// pointer, per ISA 10.2 aperture mapping). GV addressing mode (saddr = off).
__device__ __forceinline__ void async_load_b128(unsigned lds_byte_addr,
                                                const char* gaddr) {
  asm volatile("global_load_async_to_lds_b128 %0, %1, off"
               :: "v"(lds_byte_addr), "v"((unsigned long long)gaddr)
               : "memory");
}

__global__ __launch_bounds__(TPB)
void pool2x2_minmaxprod(const float* __restrict__ in,
                        float* __restrict__ out,
                        int numStages)
{
  __shared__ float smem[2 * STAGE_FLOATS];   // 48 KB double buffer (of 320 KB/WGP)
  const int t = threadIdx.x;
  // Low 32 bits of the flat shared-memory address == LDS byte offset.
  const unsigned ldsBase = (unsigned)(unsigned long long)(&smem[0]);

  const int stride = gridDim.x;
  int s = blockIdx.x;
  if (s >= numStages) return;  // uniform; grid is sized <= numStages anyway

  auto issue = [&](int stageIdx, int buf) {
    const char* g = (const char*)in + (long long)stageIdx * STAGE_BYTES + t * 16;
    const unsigned l = ldsBase + (unsigned)buf * STAGE_BYTES + (unsigned)t * 16;
#pragma unroll
    for (int k = 0; k < PAIRS_PER_STAGE; ++k)   // 192 lanes x 16 B == one row pair
      async_load_b128(l + (unsigned)k * PAIR_BYTES,
                      g + (long long)k * PAIR_BYTES);
  };

  int buf = 0;
  issue(s, 0);

  for (; s < numStages; s += stride) {
    const int snext = s + stride;
    if (snext < numStages) {
      issue(snext, buf ^ 1);                          // prefetch next stage
      asm volatile("s_wait_asynccnt 8" ::: "memory"); // in-order: first 8 done
    } else {
      asm volatile("s_wait_asynccnt 0" ::: "memory");
    }
    __syncthreads();  // stage data visible to all 6 waves

    const float* sb = smem + buf * STAGE_FLOATS;
    float res[PAIRS_PER_STAGE];
#pragma unroll
    for (int p = 0; p < PAIRS_PER_STAGE; ++p) {
      // ds_load_b64 x2: banks 0..63 each touched once per wave (conflict-free)
      const float2 a = *(const float2*)(sb + p * PAIR_FLOATS +        2 * t);
      const float2 b = *(const float2*)(sb + p * PAIR_FLOATS + W_IN + 2 * t);
      const float x0 = a.x, x1 = a.y, x2 = b.x, x3 = b.y;
      const float mn = fminf(fminf(x0, x1), fminf(x2, x3));
      const float mx = fmaxf(fmaxf(x0, x1), fmaxf(x2, x3));
      const float d  = mx - mn;
      const float inv = 1.0f / (d + EPS);
      const float prod = ((x0 - mn) * inv) * ((x1 - mn) * inv) *
                         ((x2 - mn) * inv) * ((x3 - mn) * inv);
      res[p] = prod * d + mn;   // analytically == mn; computed faithfully
    }

    // Streaming output: coalesced, non-temporal (no reuse; keep L2 for reads).
    float* ob = out + (long long)s * (PAIRS_PER_STAGE * W_OUT) + t;
#pragma unroll
    for (int p = 0; p < PAIRS_PER_STAGE; ++p)
      __builtin_nontemporal_store(res[p], ob + p * W_OUT);

    __syncthreads();  // all waves done reading 'buf' before stage s+2 overwrites it
    buf ^= 1;
  }
}

} // namespace

extern "C" void kernel_launch(void* const* d_in, const int* in_sizes, int n_in,
                              void* d_out, int out_size, void* d_ws, size_t ws_size,
                              hipStream_t stream) {
  (void)n_in; (void)out_size; (void)d_ws; (void)ws_size;
  const float* in = (const float*)d_in[0];
  float* out = (float*)d_out;

  const long long n = in_sizes[0];                       // 16*64*384*384
  const long long totalPairs = n / (2 * W_IN);           // 196,608 row pairs
  const int numStages = (int)(totalPairs / PAIRS_PER_STAGE); // 24,576 stages
  const int blocks = numStages < 4096 ? numStages : 4096;    // ~6 stages/block

  pool2x2_minmaxprod<<<blocks, TPB, 0, stream>>>(in, out, numStages);
}